// MultiHeadAttention_88948772700542
// MI455X (gfx1250) — compile-verified
//
#include <hip/hip_runtime.h>
#include <hip/hip_bf16.h>

// MI455X / gfx1250, wave32. All heavy GEMMs run on V_WMMA_F32_16X16X4_F32
// (exact f32 — matches the f32 reference, D = A(16x4) x B(4x16) + C(16x16)).
//
// Pipeline (never materializes the 134MB [B,H,N,N,D] new_edges tensor):
//   1) qkv   = nodes @ W_qkv + b_qkv                      (generic WMMA GEMM)
//   2) fused edge kernel: per wave, one (b,q,k-tile16):
//        ss GEMM (K=64) -> shift/scale tiles (WMMA)
//        new_edges = q*k*(scale+1)+shift  (elementwise, in C-fragment layout)
//        logits    = per-head sum via lane shuffles -> ws
//        leaky_relu -> LDS stage (XOR-swizzled) -> out GEMM 256->64 (WMMA)
//   3) softmax over the QUERY axis (axis=2, as in the reference) in-place
//   4) weighted = attention @ v   per (b,h)               (generic WMMA GEMM)
//   5) new_nodes = weighted @ W_nodes + b_nodes           (generic WMMA GEMM)
//
// Workspace use: (393216 + 1048576 + 131072) floats = 6 MB.

typedef __attribute__((ext_vector_type(2))) float v2f;
typedef __attribute__((ext_vector_type(8))) float v8f;

__device__ __forceinline__ v8f wmma4(v2f a, v2f b, v8f c) {
  // 8 args: (neg_a, A, neg_b, B, c_mod, C, reuse_a, reuse_b)
  return __builtin_amdgcn_wmma_f32_16x16x4_f32(false, a, false, b, (short)0, c,
                                               false, false);
}

__device__ __forceinline__ v2f ld2(const float* p) { return *(const v2f*)p; }

// ---------------------------------------------------------------------------
// Generic batched WMMA GEMM: C[M,N] = A[M,K] @ B[K,N] (+ bias), row-major,
// one wave per 16x16 C tile, 4 waves per block. Batch dims via grid y/z.
// ---------------------------------------------------------------------------
__global__ __launch_bounds__(128) void wmma_gemm_kernel(
    const float* __restrict__ A, int lda, long sAy, long sAz,
    const float* __restrict__ B, int ldb, long sBy, long sBz,
    const float* __restrict__ bias,
    float* __restrict__ C, int ldc, long sCy, long sCz,
    int M, int N, int K) {
  const int lane = threadIdx.x & 31;
  const int wave = threadIdx.x >> 5;
  const int half = lane >> 4;
  const int l16  = lane & 15;
  const int ntiles = N >> 4;
  int wt = blockIdx.x * 4 + wave;
  int mt = wt / ntiles, nt = wt % ntiles;
  if (mt * 16 >= M) return;
  A += blockIdx.y * sAy + blockIdx.z * sAz;
  B += blockIdx.y * sBy + blockIdx.z * sBz;
  C += blockIdx.y * sCy + blockIdx.z * sCz;
  const float* arow = A + (mt * 16 + l16) * (long)lda;
  const int col = nt * 16 + l16;
  v8f acc = {};
  for (int kk = 2 * half; kk < K; kk += 4) {
    v2f a = ld2(arow + kk);  // A[M=l16][kk], A[M=l16][kk+1]
    v2f b;
    b.x = B[(long)kk * ldb + col];
    b.y = B[(long)(kk + 1) * ldb + col];
    acc = wmma4(a, b, acc);
  }
  float bb = bias ? bias[col] : 0.0f;
#pragma unroll
  for (int v = 0; v < 8; ++v) {
    int row = mt * 16 + v + 8 * half;
    C[(long)row * ldc + col] = acc[v] + bb;
  }
}

// ---------------------------------------------------------------------------
// Fused edge kernel. One wave = one (b, q, k0..k0+15) tile of 16 edge rows.
// ---------------------------------------------------------------------------
// LDS staging: 16 rows x 256 feats, XOR swizzle keeps stores (16 lanes, same
// row, consecutive feats) and loads (16 rows, same feat pair) conflict-free,
// and preserves even-pair adjacency for aligned ds b64 reads.
#define STG(row, f) (((row) << 8) + ((f) ^ (((row) & 15) << 2)))

__global__ __launch_bounds__(128) void fused_edge_kernel(
    const float* __restrict__ edges,    // [B,256,256,64]
    const float* __restrict__ qkv,      // [B,256,768] (q|k|v)
    const float* __restrict__ W_ss,     // [64,512]
    const float* __restrict__ b_ss,     // [512]
    const float* __restrict__ W_edges,  // [256,64]
    const float* __restrict__ b_edges,  // [64]
    float* __restrict__ logits,         // [B,8,256,256]
    float* __restrict__ outE) {         // [B,256,256,64]
  __shared__ float stage[4][16 * 256];  // 64 KB
  const int lane = threadIdx.x & 31;
  const int wave = threadIdx.x >> 5;
  const int half = lane >> 4;
  const int l16  = lane & 15;

  const int tile = blockIdx.x * 4 + wave;  // 0..8191
  const int b  = tile >> 12;
  const int qi = (tile >> 4) & 255;
  const int k0 = (tile & 15) << 4;

  const float* Abase = edges + ((long)((b * 256 + qi) * 256 + k0) << 6);
  const float* qrow  = qkv + (long)(b * 256 + qi) * 768;        // q features
  const float* krow  = qkv + (long)(b * 256 + k0) * 768 + 256;  // k features

  // Hoist GEMM1 A fragments (edge tile, K=64 -> 16 k-steps), reused 32x.
  v2f afrag[16];
#pragma unroll
  for (int t = 0; t < 16; ++t) {
    int kk = 4 * t + 2 * half;
    afrag[t] = ld2(Abase + l16 * 64 + kk);
  }

  float racc[8];
#pragma unroll
  for (int v = 0; v < 8; ++v) racc[v] = 0.0f;

  // Phase 1: 16 feature tiles of 16 (hd = h*32+d), shift+scale GEMMs fused.
  for (int nt = 0; nt < 16; ++nt) {
    const int f0 = nt << 4;
    const int col = f0 + l16;
    v8f cs = {}, cc = {};
#pragma unroll
    for (int t = 0; t < 16; ++t) {
      int kk = 4 * t + 2 * half;
      v2f bs, bc;
      bs.x = W_ss[kk * 512 + col];
      bs.y = W_ss[(kk + 1) * 512 + col];
      bc.x = W_ss[kk * 512 + 256 + col];
      bc.y = W_ss[(kk + 1) * 512 + 256 + col];
      cs = wmma4(afrag[t], bs, cs);  // shift tile
      cc = wmma4(afrag[t], bc, cc);  // scale tile
    }
    const float shb = b_ss[col];
    const float scb = b_ss[256 + col];
    const float qf  = qrow[col];
#pragma unroll
    for (int v = 0; v < 8; ++v) {
      int row = v + 8 * half;  // k index within tile
      float kv = krow[row * 768 + col];
      float attn = qf * kv;
      float ne = fmaf(attn, cc[v] + scb + 1.0f, cs[v] + shb);
      racc[v] += ne;
      float lr = ne > 0.0f ? ne : 0.1f * ne;  // leaky_relu(0.1)
      stage[wave][STG(row, col)] = lr;
    }
    if (nt & 1) {  // head complete (2 tiles of 16 = 32 d-values)
      int head = nt >> 1;
#pragma unroll
      for (int v = 0; v < 8; ++v) {
        float s = racc[v];
        for (int m = 1; m < 16; m <<= 1) s += __shfl_xor(s, m, 32);
        racc[v] = s;
      }
      if (l16 == 0) {  // lane 0 -> rows 0..7, lane 16 -> rows 8..15
        float* lp = logits + ((long)((b * 8 + head) * 256 + qi) << 8) + k0 +
                    8 * half;
#pragma unroll
        for (int v = 0; v < 8; ++v)
          lp[v] = racc[v] * 0.17677669529663687f;  // 1/sqrt(32)
      }
#pragma unroll
      for (int v = 0; v < 8; ++v) racc[v] = 0.0f;
    }
  }

  __syncthreads();  // uniform across the 4 waves

  // Phase 2: out = leaky_relu(new_edges)[16x256] @ W_edges[256x64] + b_edges
  float* orow = outE + ((long)((b * 256 + qi) * 256 + k0) << 6);
#pragma unroll
  for (int nt2 = 0; nt2 < 4; ++nt2) {
    const int col2 = (nt2 << 4) + l16;
    v8f acc = {};
    for (int t = 0; t < 64; ++t) {
      int kk = 4 * t + 2 * half;
      v2f a = ld2(&stage[wave][STG(l16, kk)]);  // aligned ds b64
      v2f bw;
      bw.x = W_edges[kk * 64 + col2];
      bw.y = W_edges[(kk + 1) * 64 + col2];
      acc = wmma4(a, bw, acc);
    }
    float bb = b_edges[col2];
#pragma unroll
    for (int v = 0; v < 8; ++v) {
      int row = v + 8 * half;
      orow[(row << 6) + col2] = acc[v] + bb;
    }
  }
}

// ---------------------------------------------------------------------------
// Softmax over the QUERY axis (axis=2): column softmax, one thread per
// (b,h,k) column, strided reads are coalesced across adjacent k.
// ---------------------------------------------------------------------------
__global__ __launch_bounds__(256) void softmax_q_kernel(
    float* __restrict__ logits) {  // [B*8, 256(q), 256(k)], in-place
  int c = blockIdx.x * 256 + threadIdx.x;  // 0..4095
  int bh = c >> 8;
  int k  = c & 255;
  float* base = logits + ((long)bh << 16) + k;
  float m = -3.4e38f, s = 0.0f;
  for (int q = 0; q < 256; ++q) {  // online max/sum
    float x = base[q << 8];
    float nm = fmaxf(m, x);
    s = s * __expf(m - nm) + __expf(x - nm);
    m = nm;
  }
  float inv = 1.0f / s;
  for (int q = 0; q < 256; ++q) {
    float x = base[q << 8];
    base[q << 8] = __expf(x - m) * inv;
  }
}

extern "C" void kernel_launch(void* const* d_in, const int* in_sizes, int n_in,
                              void* d_out, int out_size, void* d_ws,
                              size_t ws_size, hipStream_t stream) {
  const float* nodes   = (const float*)d_in[0];  // [2,256,256]
  const float* edges   = (const float*)d_in[1];  // [2,256,256,64]
  const float* W_qkv   = (const float*)d_in[2];  // [256,768]
  const float* b_qkv   = (const float*)d_in[3];  // [768]
  const float* W_ss    = (const float*)d_in[4];  // [64,512]
  const float* b_ss    = (const float*)d_in[5];  // [512]
  const float* W_nodes = (const float*)d_in[6];  // [256,256]
  const float* b_nodes = (const float*)d_in[7];  // [256]
  const float* W_edges = (const float*)d_in[8];  // [256,64]
  const float* b_edges = (const float*)d_in[9];  // [64]

  float* out_nodes = (float*)d_out;              // [2,256,256]
  float* out_edges = out_nodes + 131072;         // [2,256,256,64]

  float* qkv_ws      = (float*)d_ws;             // 393216 f
  float* logits_ws   = qkv_ws + 393216;          // 1048576 f (logits->attn)
  float* weighted_ws = logits_ws + 1048576;      // 131072 f
  (void)in_sizes; (void)n_in; (void)out_size; (void)ws_size;

  dim3 blk(128);

  // 1) qkv = nodes @ W_qkv + b_qkv : [512x256]@[256x768]
  wmma_gemm_kernel<<<dim3((32 * 48) / 4, 1, 1), blk, 0, stream>>>(
      nodes, 256, 0, 0, W_qkv, 768, 0, 0, b_qkv, qkv_ws, 768, 0, 0, 512, 768,
      256);

  // 2) fused edge pipeline (8192 wave-tiles, 4 per block)
  fused_edge_kernel<<<dim3(2048), blk, 0, stream>>>(
      edges, qkv_ws, W_ss, b_ss, W_edges, b_edges, logits_ws, out_edges);

  // 3) softmax over q (axis=2), in place
  softmax_q_kernel<<<dim3(16), dim3(256), 0, stream>>>(logits_ws);

  // 4) weighted[b,q,h*32+d] = sum_k attn[b,h,q,k] * v[b,k,512+h*32+d]
  //    per (b,h): [256x256]@[256x32]; grid.y = h (8), grid.z = b (2)
  wmma_gemm_kernel<<<dim3(8, 8, 2), blk, 0, stream>>>(
      logits_ws, 256, 65536L, 524288L, qkv_ws + 512, 768, 32L, 196608L,
      nullptr, weighted_ws, 256, 32L, 65536L, 256, 32, 256);

  // 5) new_nodes = weighted @ W_nodes + b_nodes : [512x256]@[256x256]
  wmma_gemm_kernel<<<dim3((32 * 16) / 4, 1, 1), blk, 0, stream>>>(
      weighted_ws, 256, 0, 0, W_nodes, 256, 0, 0, b_nodes, out_nodes, 256, 0,
      0, 512, 256, 256);
}